// graphConvPool_11141145166097
// MI455X (gfx1250) — compile-verified
//
#include <hip/hip_runtime.h>
#include <hip/hip_bf16.h>
#include <stdint.h>

#define NN      100000
#define INDIM   512
#define EE      3200000
#define KP      10000
#define NBINS   4096
#define BIN_SHIFT 20
#define SORT_N  16384

#define ROWS_PB 128
#define KTILE   64
#define KTILES  (INDIM / KTILE)          // 8
#define TILE_F  (ROWS_PB * KTILE)        // 8192 floats = 32KB

typedef float v2f __attribute__((ext_vector_type(2)));
typedef float v8f __attribute__((ext_vector_type(8)));

// ---------------------------------------------------------------------------
// 1) GEMV h = x @ W: async global->LDS tile staging (double buffered) feeding
//    V_WMMA_F32_16X16X4_F32. 128 rows/block, K tiled by 64.
// ---------------------------------------------------------------------------
__global__ void __launch_bounds__(256)
gemv_wmma_kernel(const float* __restrict__ x, const float* __restrict__ W,
                 float* __restrict__ h, int n) {
    __shared__ float sW[INDIM];          // 2KB
    __shared__ float sX[2][TILE_F];      // 2 x 32KB

    const int tid  = threadIdx.x;
    const int wave = tid >> 5;
    const int lane = tid & 31;
    const int row0 = blockIdx.x * ROWS_PB;

    // stage W (512 floats, 2 per thread)
    sW[tid]       = W[tid];
    sW[tid + 256] = W[tid + 256];

    // one tile = 2048 float4 async copies; 8 instructions per thread
    auto issue_tile = [&](int t, int buf) {
        const int k0 = t * KTILE;
#pragma unroll
        for (int i = 0; i < 8; ++i) {
            const int fe = i * 256 + tid;           // float4 slot 0..2047
            const int r  = fe >> 4;                 // tile row 0..127
            const int c4 = fe & 15;                 // float4 col 0..15
            const int gr = min(row0 + r, n - 1);    // clamp: never read past x
            const float*   gp = x + (size_t)gr * INDIM + k0 + c4 * 4;
            const uint32_t lo = (uint32_t)(uintptr_t)(&sX[buf][r * KTILE + c4 * 4]);
            asm volatile("global_load_async_to_lds_b128 %0, %1, off"
                         :: "v"(lo), "v"(gp) : "memory");
        }
    };

    issue_tile(0, 0);

    const int laneRow = lane & 15;
    const int kSel    = (lane >> 4) << 1;  // lanes 0-15 -> K=0,1 ; 16-31 -> K=2,3

    v8f c = {};
    for (int t = 0; t < KTILES; ++t) {
        const int buf = t & 1;
        if (t + 1 < KTILES) {
            issue_tile(t + 1, buf ^ 1);             // prefetch next tile
            asm volatile("s_wait_asynccnt 0x8" ::: "memory");  // tile t arrived
        } else {
            asm volatile("s_wait_asynccnt 0x0" ::: "memory");
        }
        __syncthreads();                            // tile t visible block-wide
        const float2* px = reinterpret_cast<const float2*>(
            &sX[buf][(wave * 16 + laneRow) * KTILE]);
        const float2* pw = reinterpret_cast<const float2*>(&sW[t * KTILE]);
#pragma unroll
        for (int kk = 0; kk < KTILE; kk += 4) {
            float2 av = px[(kk + kSel) >> 1];
            float2 bv = pw[(kk + kSel) >> 1];
            v2f a; a[0] = av.x; a[1] = av.y;
            v2f b; b[0] = bv.x; b[1] = bv.y;
            c = __builtin_amdgcn_wmma_f32_16x16x4_f32(
                    /*neg_a=*/false, a, /*neg_b=*/false, b,
                    /*c_mod=*/(short)0, c, /*reuse_a=*/false, /*reuse_b=*/false);
        }
        __syncthreads();                            // done reading buf before reuse
    }

    // column N=0 of C: lane 0 holds M=0..7, lane 16 holds M=8..15 (VGPRs 0..7)
    if ((lane & 15) == 0) {
        const int mbase = row0 + wave * 16 + (lane >> 4) * 8;
#pragma unroll
        for (int t2 = 0; t2 < 8; ++t2) {
            const int r = mbase + t2;
            if (r < n) h[r] = c[t2];
        }
    }
}

// ---------------------------------------------------------------------------
// 2) init: deg=1 (self loop), pos=-1, hist=0, counter=0
// ---------------------------------------------------------------------------
__global__ void init_kernel(float* deg, int* pos, int* hist, int* counter, int n) {
    const int i = blockIdx.x * blockDim.x + threadIdx.x;
    if (i < n) { deg[i] = 1.0f; pos[i] = -1; }
    if (i < NBINS) hist[i] = 0;
    if (i == 0) *counter = 0;
}

// 3) degree accumulation over edge destinations
__global__ void edge_deg_kernel(const int* __restrict__ dst, float* __restrict__ deg, int e) {
    const int i = blockIdx.x * blockDim.x + threadIdx.x;
    if (i < e) atomicAdd(&deg[dst[i]], 1.0f);
}

// 4) dinv = rsqrt(deg); score = dinv^2 * h (self-loop term) + b
__global__ void node_score_kernel(const float* __restrict__ deg, const float* __restrict__ h,
                                  const float* __restrict__ b, float* __restrict__ dinv,
                                  float* __restrict__ score, int n) {
    const int i = blockIdx.x * blockDim.x + threadIdx.x;
    if (i >= n) return;
    const float d  = deg[i];
    const float di = (d > 0.0f) ? rsqrtf(d) : 0.0f;
    dinv[i]  = di;
    score[i] = di * di * h[i] + b[0];
}

// 5) score[dst] += dinv[src]*dinv[dst]*h[src]
__global__ void edge_scatter_kernel(const int* __restrict__ src, const int* __restrict__ dst,
                                    const float* __restrict__ dinv, const float* __restrict__ h,
                                    float* __restrict__ score, int e) {
    const int i = blockIdx.x * blockDim.x + threadIdx.x;
    if (i >= e) return;
    const int s = src[i], d = dst[i];
    atomicAdd(&score[d], dinv[s] * dinv[d] * h[s]);
}

// 6) sortable key (flip float bits) + 4096-bin histogram of top 12 bits
__global__ void key_hist_kernel(const float* __restrict__ score, uint32_t* __restrict__ key,
                                int* __restrict__ hist, int n) {
    const int i = blockIdx.x * blockDim.x + threadIdx.x;
    if (i >= n) return;
    uint32_t u = __float_as_uint(score[i]);
    u ^= (u & 0x80000000u) ? 0xFFFFFFFFu : 0x80000000u;
    key[i] = u;
    atomicAdd(&hist[u >> BIN_SHIFT], 1);
}

// 7) scan histogram high->low for threshold bin (cumulative >= KP)
__global__ void threshold_kernel(const int* __restrict__ hist, int* __restrict__ tbin) {
    if (threadIdx.x == 0 && blockIdx.x == 0) {
        int cum = 0, t = 0;
        for (int b = NBINS - 1; b >= 0; --b) {
            cum += hist[b];
            if (cum >= KP) { t = b; break; }
        }
        *tbin = t;
    }
}

// 8) compact candidates whose bin >= threshold
__global__ void compact_kernel(const uint32_t* __restrict__ key, const int* __restrict__ tbin,
                               uint32_t* __restrict__ candKey, int* __restrict__ candIdx,
                               int* __restrict__ counter, int n) {
    const int i = blockIdx.x * blockDim.x + threadIdx.x;
    if (i >= n) return;
    const int T = *tbin;
    if ((int)(key[i] >> BIN_SHIFT) >= T) {
        const int p = atomicAdd(counter, 1);
        if (p < SORT_N) { candKey[p] = key[i]; candIdx[p] = i; }
    }
}

// 9) one-workgroup bitonic sort of 16384 (key,idx) pairs in 128KB LDS, descending
__global__ void __launch_bounds__(1024)
sort_emit_kernel(const uint32_t* __restrict__ candKey, const int* __restrict__ candIdx,
                 const int* __restrict__ counter, int* __restrict__ selIdx,
                 float* __restrict__ outIdx) {
    __shared__ uint32_t skey[SORT_N];
    __shared__ int      sidx[SORT_N];
    const int tid = threadIdx.x;
    int count = *counter; if (count > SORT_N) count = SORT_N;
    for (int i = tid; i < SORT_N; i += 1024) {
        if (i < count) { skey[i] = candKey[i]; sidx[i] = candIdx[i]; }
        else           { skey[i] = 0u;         sidx[i] = 0x7FFFFFFF; }
    }
    __syncthreads();
    for (int k = 2; k <= SORT_N; k <<= 1) {
        for (int j = k >> 1; j > 0; j >>= 1) {
            for (int i = tid; i < SORT_N; i += 1024) {
                const int ixj = i ^ j;
                if (ixj > i) {
                    const bool up = ((i & k) == 0);
                    const uint32_t ka = skey[i],   kb = skey[ixj];
                    const int      ia = sidx[i],   ib = sidx[ixj];
                    const bool aGt = (ka > kb) || (ka == kb && ia < ib);
                    const bool doSwap = up ? !aGt : aGt;   // descending overall
                    if (doSwap) {
                        skey[i] = kb; sidx[i] = ib;
                        skey[ixj] = ka; sidx[ixj] = ia;
                    }
                }
            }
            __syncthreads();
        }
    }
    for (int r = tid; r < KP; r += 1024) {
        selIdx[r] = sidx[r];
        outIdx[r] = (float)sidx[r];
    }
}

// 10) pos[selIdx[r]] = r
__global__ void set_pos_kernel(const int* __restrict__ selIdx, int* __restrict__ pos) {
    const int r = blockIdx.x * blockDim.x + threadIdx.x;
    if (r < KP) pos[selIdx[r]] = r;
}

// 11) new_g = x[idx]  (float4 row copies)
__global__ void __launch_bounds__(128)
gather_kernel(const float* __restrict__ x, const int* __restrict__ selIdx,
              float* __restrict__ outG) {
    const int row = blockIdx.x;
    const size_t s = (size_t)selIdx[row];
    const float4* in  = reinterpret_cast<const float4*>(x + s * INDIM);
    float4*       out = reinterpret_cast<float4*>(outG + (size_t)row * INDIM);
    out[threadIdx.x] = in[threadIdx.x];
}

// 12) zero the 400MB pooled adjacency (float4 stores — the HBM-dominant step)
__global__ void fill_zero4_kernel(float4* __restrict__ p, size_t n4) {
    const size_t i = (size_t)blockIdx.x * blockDim.x + threadIdx.x;
    if (i < n4) p[i] = make_float4(0.f, 0.f, 0.f, 0.f);
}

// 13) adj[pu,pv] += 1 for surviving edges
__global__ void adj_edges_kernel(const int* __restrict__ src, const int* __restrict__ dst,
                                 const int* __restrict__ pos, float* __restrict__ adj, int e) {
    const int i = blockIdx.x * blockDim.x + threadIdx.x;
    if (i >= e) return;
    const int pu = pos[src[i]];
    const int pv = pos[dst[i]];
    if (pu >= 0 && pv >= 0) atomicAdd(&adj[(size_t)pu * KP + pv], 1.0f);
}

// ---------------------------------------------------------------------------
extern "C" void kernel_launch(void* const* d_in, const int* in_sizes, int n_in,
                              void* d_out, int out_size, void* d_ws, size_t ws_size,
                              hipStream_t stream) {
    const float* x  = (const float*)d_in[0];
    const int*   ei = (const int*)d_in[1];          // [2, E] row-major
    const float* W  = (const float*)d_in[2];
    const float* b  = (const float*)d_in[3];
    const int* src = ei;
    const int* dst = ei + EE;

    float* out    = (float*)d_out;
    float* outG   = out;                                  // [KP, 512]
    float* outAdj = out + (size_t)KP * INDIM;             // [KP, KP]
    float* outIdx = outAdj + (size_t)KP * KP;             // [KP] (as float)

    // workspace layout (4-byte units)
    float*    h       = (float*)d_ws;
    float*    deg     = h + NN;
    float*    dinv    = deg + NN;
    float*    score   = dinv + NN;
    uint32_t* key     = (uint32_t*)(score + NN);
    int*      pos     = (int*)(key + NN);
    int*      hist    = pos + NN;
    int*      counter = hist + NBINS;
    int*      tbin    = counter + 1;
    uint32_t* candKey = (uint32_t*)(tbin + 7);            // aligned spare
    int*      candIdx = (int*)(candKey + SORT_N);
    int*      selIdx  = candIdx + SORT_N;

    const int TB = 256;
    const int gN = (NN + TB - 1) / TB;
    const int gE = (EE + TB - 1) / TB;

    init_kernel<<<gN, TB, 0, stream>>>(deg, pos, hist, counter, NN);
    gemv_wmma_kernel<<<(NN + ROWS_PB - 1) / ROWS_PB, 256, 0, stream>>>(x, W, h, NN);
    edge_deg_kernel<<<gE, TB, 0, stream>>>(dst, deg, EE);
    node_score_kernel<<<gN, TB, 0, stream>>>(deg, h, b, dinv, score, NN);
    edge_scatter_kernel<<<gE, TB, 0, stream>>>(src, dst, dinv, h, score, EE);
    key_hist_kernel<<<gN, TB, 0, stream>>>(score, key, hist, NN);
    threshold_kernel<<<1, 32, 0, stream>>>(hist, tbin);
    compact_kernel<<<gN, TB, 0, stream>>>(key, tbin, candKey, candIdx, counter, NN);
    sort_emit_kernel<<<1, 1024, 0, stream>>>(candKey, candIdx, counter, selIdx, outIdx);
    set_pos_kernel<<<(KP + TB - 1) / TB, TB, 0, stream>>>(selIdx, pos);
    gather_kernel<<<KP, 128, 0, stream>>>(x, selIdx, outG);
    {
        const size_t n4 = ((size_t)KP * KP) / 4;          // 25M float4
        const int gZ = (int)((n4 + TB - 1) / TB);
        fill_zero4_kernel<<<gZ, TB, 0, stream>>>((float4*)outAdj, n4);
    }
    adj_edges_kernel<<<gE, TB, 0, stream>>>(src, dst, pos, outAdj, EE);
}